// BasicNetwork_75230647157379
// MI455X (gfx1250) — compile-verified
//
#include <hip/hip_runtime.h>
#include <hip/hip_bf16.h>

#define N_NODES 100000
#define N_EDGES 1000000

typedef float v2f __attribute__((ext_vector_type(2)));
typedef float v8f __attribute__((ext_vector_type(8)));

// ---------------------------------------------------------------------------
// Degree / normalization kernels
// ---------------------------------------------------------------------------
__global__ void zero_f32_kernel(float* __restrict__ p, int n) {
    int i = blockIdx.x * blockDim.x + threadIdx.x;
    if (i < n) p[i] = 0.0f;
}

__global__ void count_deg_kernel(const int* __restrict__ dst,
                                 float* __restrict__ deg, int e) {
    int i = blockIdx.x * blockDim.x + threadIdx.x;
    if (i < e) atomicAdd(&deg[dst[i]], 1.0f);
}

// dinv[v] = rsqrt(deg[v] + 1)   (self-loop contributes 1; always > 0)
__global__ void make_dinv_kernel(float* __restrict__ deg, int n) {
    int i = blockIdx.x * blockDim.x + threadIdx.x;
    if (i < n) deg[i] = rsqrtf(deg[i] + 1.0f);
}

// ---------------------------------------------------------------------------
// Dense GEMM via V_WMMA_F32_16X16X4_F32 (exact fp32 math).
// One wave computes one 16(M) x 16(N) tile; block = (DOUT/16) waves, each
// wave owns one 16-wide column tile. Grid.x = N/16 row tiles.
//
// VGPR layouts per CDNA5 ISA 7.12.2 (wave32):
//   A 16x4 f32 : lanes 0-15 -> M=lane,K={0,1}; lanes 16-31 -> M=lane-16,K={2,3}
//   B 4x16 f32 : lanes 0-15 -> N=lane,K={0,1}; lanes 16-31 -> N=lane-16,K={2,3}
//   C/D 16x16  : VGPR r: lanes 0-15 -> M=r, lanes 16-31 -> M=r+8; N=lane&15
// ---------------------------------------------------------------------------
template <int DIN, int DOUT>
__global__ void gemm_wmma_kernel(const float* __restrict__ X,
                                 const float* __restrict__ W,
                                 float* __restrict__ H) {
    const int lane  = threadIdx.x & 31;
    const int wave  = threadIdx.x >> 5;
    const int rowBase = blockIdx.x * 16;
    const int colBase = wave * 16;
    const int l15   = lane & 15;
    const int khalf = lane >> 4;          // 0 or 1 -> K pair {0,1} or {2,3}

    v8f acc = {};
    const float* a_ptr = X + (size_t)(rowBase + l15) * DIN + khalf * 2;
    const float* b_col = W + colBase + l15;

#pragma unroll
    for (int kk = 0; kk < DIN; kk += 4) {
        v2f a, b;
        a.x = a_ptr[kk + 0];
        a.y = a_ptr[kk + 1];
        b.x = b_col[(size_t)(kk + khalf * 2 + 0) * DOUT];
        b.y = b_col[(size_t)(kk + khalf * 2 + 1) * DOUT];
        acc = __builtin_amdgcn_wmma_f32_16x16x4_f32(
            /*neg_a=*/false, a, /*neg_b=*/false, b,
            /*c_mod=*/(short)0, acc, /*reuse_a=*/false, /*reuse_b=*/false);
    }

    float* hbase = H + (size_t)rowBase * DOUT + colBase;
#pragma unroll
    for (int r = 0; r < 8; ++r) {
        int m = r + 8 * khalf;
        hbase[(size_t)m * DOUT + l15] = acc[r];
    }
}

// ---------------------------------------------------------------------------
// Self-loop init: AGG[v][c] = dinv[v]^2 * H[v][c]
// ---------------------------------------------------------------------------
__global__ void agg_init_kernel(const float* __restrict__ H,
                                const float* __restrict__ dinv,
                                float* __restrict__ AGG,
                                int total, int log2d) {
    int i = blockIdx.x * blockDim.x + threadIdx.x;
    if (i >= total) return;
    int v = i >> log2d;
    float dv = dinv[v];
    AGG[i] = dv * dv * H[i];
}

// ---------------------------------------------------------------------------
// Edge scatter: AGG[dst][c] += dinv[src]*dinv[dst] * H[src][c]
// One thread per (edge, 4-channel quad); float4 gather, 4 f32 atomics (L2-hot).
// ---------------------------------------------------------------------------
__global__ void edge_scatter_kernel(const int* __restrict__ src,
                                    const int* __restrict__ dst,
                                    const float* __restrict__ dinv,
                                    const float* __restrict__ H,
                                    float* __restrict__ AGG,
                                    int total, int log2q, int log2d) {
    int tid = blockIdx.x * blockDim.x + threadIdx.x;
    if (tid >= total) return;
    int e = tid >> log2q;
    int c = (tid & ((1 << log2q) - 1)) * 4;
    int s = src[e];
    int d = dst[e];
    float norm = dinv[s] * dinv[d];
    const float4 hv = *(const float4*)(H + ((size_t)s << log2d) + c);
    float* ab = AGG + ((size_t)d << log2d) + c;
    atomicAdd(ab + 0, norm * hv.x);
    atomicAdd(ab + 1, norm * hv.y);
    atomicAdd(ab + 2, norm * hv.z);
    atomicAdd(ab + 3, norm * hv.w);
}

// ---------------------------------------------------------------------------
// OUT[v][c] = (relu?) (IN[v][c] + b[c])
// ---------------------------------------------------------------------------
__global__ void bias_act_kernel(const float* __restrict__ IN,
                                const float* __restrict__ b,
                                float* __restrict__ OUT,
                                int total, int cmask, int relu) {
    int i = blockIdx.x * blockDim.x + threadIdx.x;
    if (i >= total) return;
    float v = IN[i] + b[i & cmask];
    if (relu) v = fmaxf(v, 0.0f);
    OUT[i] = v;
}

// ---------------------------------------------------------------------------
extern "C" void kernel_launch(void* const* d_in, const int* in_sizes, int n_in,
                              void* d_out, int out_size, void* d_ws, size_t ws_size,
                              hipStream_t stream) {
    const float* x   = (const float*)d_in[0];
    const int*   ei  = (const int*)d_in[1];
    const float* W1  = (const float*)d_in[2];
    const float* b1  = (const float*)d_in[3];
    const float* W2  = (const float*)d_in[4];
    const float* b2  = (const float*)d_in[5];
    const float* W3  = (const float*)d_in[6];
    const float* b3  = (const float*)d_in[7];
    float*       out = (float*)d_out;

    const int* src = ei;            // edge_index[0]
    const int* dst = ei + N_EDGES;  // edge_index[1]

    // Workspace layout (floats): dinv[100096 pad] | h[N*64] | agg[N*64]
    float* dinv = (float*)d_ws;
    float* h    = dinv + 100096;
    float* agg  = h + (size_t)N_NODES * 64;

    const int BT = 256;
    const int gN   = (N_NODES + BT - 1) / BT;
    const int gE   = (N_EDGES + BT - 1) / BT;
    const int g64  = (N_NODES * 64 + BT - 1) / BT;       // 25000
    const int g32  = (N_NODES * 32 + BT - 1) / BT;       // 12500
    const int gEs64 = (N_EDGES * 16 + BT - 1) / BT;      // 62500  (64ch / 4)
    const int gEs32 = (N_EDGES * 8  + BT - 1) / BT;      // 31250  (32ch / 4)
    const int rowTiles = N_NODES / 16;                   // 6250

    // ---- normalization (graph-only, computed once) ----
    zero_f32_kernel<<<gN, BT, 0, stream>>>(dinv, N_NODES);
    count_deg_kernel<<<gE, BT, 0, stream>>>(dst, dinv, N_EDGES);
    make_dinv_kernel<<<gN, BT, 0, stream>>>(dinv, N_NODES);

    // ---- Layer 1: x[N,64] @ W1[64,64] ----
    gemm_wmma_kernel<64, 64><<<rowTiles, 128, 0, stream>>>(x, W1, h);
    agg_init_kernel<<<g64, BT, 0, stream>>>(h, dinv, agg, N_NODES * 64, 6);
    edge_scatter_kernel<<<gEs64, BT, 0, stream>>>(src, dst, dinv, h, agg,
                                                  N_EDGES * 16, 4, 6);
    bias_act_kernel<<<g64, BT, 0, stream>>>(agg, b1, agg, N_NODES * 64, 63, 1);

    // ---- Layer 2: agg[N,64] @ W2[64,64] ----
    gemm_wmma_kernel<64, 64><<<rowTiles, 128, 0, stream>>>(agg, W2, h);
    agg_init_kernel<<<g64, BT, 0, stream>>>(h, dinv, agg, N_NODES * 64, 6);
    edge_scatter_kernel<<<gEs64, BT, 0, stream>>>(src, dst, dinv, h, agg,
                                                  N_EDGES * 16, 4, 6);
    bias_act_kernel<<<g64, BT, 0, stream>>>(agg, b2, agg, N_NODES * 64, 63, 1);

    // ---- Layer 3: agg[N,64] @ W3[64,32] -> d_out ----
    gemm_wmma_kernel<64, 32><<<rowTiles, 64, 0, stream>>>(agg, W3, h);
    // (gemm has fully consumed agg before agg_init reuses it at width 32)
    agg_init_kernel<<<g32, BT, 0, stream>>>(h, dinv, agg, N_NODES * 32, 5);
    edge_scatter_kernel<<<gEs32, BT, 0, stream>>>(src, dst, dinv, h, agg,
                                                  N_EDGES * 8, 3, 5);
    bias_act_kernel<<<g32, BT, 0, stream>>>(agg, b3, out, N_NODES * 32, 31, 0);
}